// ConvDicoLearnFISTANN_2740189135074
// MI455X (gfx1250) — compile-verified
//
#include <hip/hip_runtime.h>
#include <cmath>
#include <cstdint>

// ---------------- problem dims ----------------
#define NCq   8
#define NZq   4
#define NTq   12
#define NUq   192
#define NFq   192
#define NFILq 8
#define PLANE 36864L                 // 192*192
#define IMGN  1769472L               // NZ*NT*PLANE
#define COILN 14155776L              // NC*NZ*NT*PLANE
#define SSPN  14155776L              // NZ*NFIL*NT*PLANE
#define PINN  32768                  // NFIL*16^3
#define PIIMG 4096                   // 16^3

typedef float v2f __attribute__((ext_vector_type(2)));
typedef float v8f __attribute__((ext_vector_type(8)));
typedef unsigned int u32x4 __attribute__((ext_vector_type(4)));
typedef int i32x4 __attribute__((ext_vector_type(4)));
typedef int i32x8 __attribute__((ext_vector_type(8)));

#if __has_builtin(__builtin_amdgcn_tensor_load_to_lds)
#define HAVE_TDM 1
#else
#define HAVE_TDM 0
#endif

// ---------------- DFT matrix build (ortho, forward sign) ----------------
__global__ void k_fill_dft(float* __restrict__ Fre, float* __restrict__ Fim) {
  int i = blockIdx.x * 256 + threadIdx.x;
  if (i >= 192 * 192) return;
  int j = i / 192, k = i % 192;
  int m = (j * k) % 192;
  double ang = -2.0 * 3.14159265358979323846 * (double)m / 192.0;
  double s = 0.07216878364870322828;  // 1/sqrt(192)
  Fre[i] = (float)(cos(ang) * s);
  Fim[i] = (float)(sin(ang) * s);
}

#if HAVE_TDM
// Issue a 2-D TDM tile load (tile_x contiguous dwords per row, tile_y rows,
// row stride stride_elems dwords) from gaddr into LDS byte offset lds_addr.
// D# layout per CDNA5 ISA ch.8 (group0 128b, group1 256b; groups 2/3 zero = 2D).
__device__ __forceinline__ void tdm_load_2d(unsigned lds_addr, const float* gaddr,
                                            int tile_x, int tile_y, int stride_elems) {
  unsigned long long ga = (unsigned long long)(uintptr_t)gaddr;
  u32x4 g0;
  g0.x = 1u;                                              // count=1 (valid), user mode
  g0.y = lds_addr;                                        // lds_addr [63:32]
  g0.z = (unsigned)(ga & 0xFFFFFFFFu);                    // global_addr low
  g0.w = (unsigned)((ga >> 32) & 0x1FFFFFFu) | (2u << 30); // addr hi | type=2
  i32x8 g1;
  g1[0] = (2 << 16);                                      // data_size=4B
  g1[1] = (int)(((unsigned)tile_x & 0xFFFFu) << 16);      // tensor_dim0[15:0]  (== tile_x)
  g1[2] = (int)(((unsigned)tile_y & 0xFFFFu) << 16);      // td0 hi=0 | tensor_dim1[15:0]
  g1[3] = (int)(((unsigned)tile_x & 0xFFFFu) << 16);      // td1 hi=0 | tile_dim0
  g1[4] = (int)((unsigned)tile_y & 0xFFFFu);              // tile_dim1 | tile_dim2=0
  g1[5] = stride_elems;                                   // tensor_dim0_stride low32
  g1[6] = 0;                                              // stride hi | td1stride lo
  g1[7] = 0;
  i32x4 z4 = {0, 0, 0, 0};
#if __clang_major__ >= 23
  i32x8 z8 = {0, 0, 0, 0, 0, 0, 0, 0};
  __builtin_amdgcn_tensor_load_to_lds(g0, g1, z4, z4, z8, 0);
#else
  __builtin_amdgcn_tensor_load_to_lds(g0, g1, z4, z4, 0);
#endif
}
#endif

// ---------------- complex GEMM C = A*B (192x192x192, batched) via f32 WMMA ------
// One 128-thread block per (image, tn): stages the 192x16 B-panel (re+im) in LDS
// (TDM when available), 4 waves x 3 tm-tiles each. aImSign/bImSign conjugate the
// DFT operand for the inverse transform.
__global__ __launch_bounds__(128) void k_cgemm(
    const float* __restrict__ Are, const float* __restrict__ Aim,
    const float* __restrict__ Bre, const float* __restrict__ Bim,
    float* __restrict__ Cre, float* __restrict__ Cim,
    long aS, long bS, long cS, float aImSign, float bImSign) {
  constexpr int N = 192;
  __shared__ float sB[2][192 * 16];  // [plane][k*16 + n] : 24 KB
  int w = threadIdx.x >> 5, lane = threadIdx.x & 31;
  int img = blockIdx.x / 12, tn = blockIdx.x % 12;
  int lo = lane & 15, hi = lane >> 4;

  const float* bpr = Bre + (long)img * bS + tn * 16;
  const float* bpi = Bim + (long)img * bS + tn * 16;

#if HAVE_TDM
  if (w == 0) {
    tdm_load_2d((unsigned)(uintptr_t)(void*)&sB[0][0], bpr, 16, 192, N);
    tdm_load_2d((unsigned)(uintptr_t)(void*)&sB[1][0], bpi, 16, 192, N);
    __builtin_amdgcn_s_wait_tensorcnt(0);
  }
#else
  for (int e = threadIdx.x; e < 192 * 16; e += 128) {
    int r = e >> 4, c = e & 15;
    sB[0][e] = bpr[(long)r * N + c];
    sB[1][e] = bpi[(long)r * N + c];
  }
#endif
  __syncthreads();

  // three tm tiles per wave: tm = w, w+4, w+8
  const float* arA[3];
  const float* aiA[3];
#pragma unroll
  for (int t = 0; t < 3; t++) {
    int tm = w + 4 * t;
    arA[t] = Are + (long)img * aS + (long)(tm * 16 + lo) * N + 2 * hi;
    aiA[t] = Aim + (long)img * aS + (long)(tm * 16 + lo) * N + 2 * hi;
  }

  v8f cr[3], ci[3];
#pragma unroll
  for (int t = 0; t < 3; t++) {
    cr[t] = {0.f, 0.f, 0.f, 0.f, 0.f, 0.f, 0.f, 0.f};
    ci[t] = {0.f, 0.f, 0.f, 0.f, 0.f, 0.f, 0.f, 0.f};
  }

  for (int k0 = 0; k0 < N; k0 += 4) {
    int idx = (k0 + 2 * hi) * 16 + lo;
    v2f b_re = {sB[0][idx], sB[0][idx + 16]};
    v2f b_im = {sB[1][idx] * bImSign, sB[1][idx + 16] * bImSign};
    __builtin_prefetch(arA[0] + k0 + 64, 0, 3);
#pragma unroll
    for (int t = 0; t < 3; t++) {
      v2f a_re = {arA[t][k0], arA[t][k0 + 1]};
      v2f a_im = {aiA[t][k0] * aImSign, aiA[t][k0 + 1] * aImSign};
      v2f a_imn = {-a_im.x, -a_im.y};
      // Cre += Are*Bre - Aim*Bim ; Cim += Are*Bim + Aim*Bre
      cr[t] = __builtin_amdgcn_wmma_f32_16x16x4_f32(false, a_re, false, b_re, (short)0, cr[t], false, false);
      cr[t] = __builtin_amdgcn_wmma_f32_16x16x4_f32(false, a_imn, false, b_im, (short)0, cr[t], false, false);
      ci[t] = __builtin_amdgcn_wmma_f32_16x16x4_f32(false, a_re, false, b_im, (short)0, ci[t], false, false);
      ci[t] = __builtin_amdgcn_wmma_f32_16x16x4_f32(false, a_im, false, b_re, (short)0, ci[t], false, false);
    }
  }
#pragma unroll
  for (int t = 0; t < 3; t++) {
    int tm = w + 4 * t;
    float* crp = Cre + (long)img * cS + (long)(tm * 16 + hi * 8) * N + tn * 16 + lo;
    float* cip = Cim + (long)img * cS + (long)(tm * 16 + hi * 8) * N + tn * 16 + lo;
#pragma unroll
    for (int r = 0; r < 8; r++) {
      crp[(long)r * N] = cr[t][r];
      cip[(long)r * N] = ci[t][r];
    }
  }
}

// ---------------- 3D circular conv: DH (correlation, per-filter out) -----------
template <int NTt, int NUt, int NFt>
__global__ __launch_bounds__(256) void k_convDH(
    const float* __restrict__ xRe, const float* __restrict__ xIm,
    const float* __restrict__ w,
    float* __restrict__ sRe, float* __restrict__ sIm) {
  constexpr int TF = NFt / 16;
  __shared__ float tile[22 * 22 * NTt];
  __shared__ float wl[NFILq * 343];
  int z = blockIdx.z;
  const float* xp = (blockIdx.y ? xIm : xRe) + (long)z * NTt * NUt * NFt;
  float* sp = (blockIdx.y ? sIm : sRe);
  int u0 = (blockIdx.x / TF) * 16, f0 = (blockIdx.x % TF) * 16;
  for (int e = threadIdx.x; e < NFILq * 343; e += 256) wl[e] = w[e];
  constexpr int TOT = 22 * 22 * NTt;
  for (int e = threadIdx.x; e < TOT; e += 256) {
    int hu = e / (22 * NTt);
    int r = e - hu * 22 * NTt;
    int hf = r / NTt, ht = r - (r / NTt) * NTt;
    int gu = u0 + hu - 3; gu += (gu < 0) ? NUt : 0; gu -= (gu >= NUt) ? NUt : 0;
    int gf = f0 + hf - 3; gf += (gf < 0) ? NFt : 0; gf -= (gf >= NFt) ? NFt : 0;
    tile[e] = xp[((long)ht * NUt + gu) * NFt + gf];
  }
  __syncthreads();
  int tu = threadIdx.x >> 4, tf = threadIdx.x & 15;
  int u = u0 + tu, f = f0 + tf;
  for (int fil = 0; fil < NFILq; ++fil) {
    float acc[NTt];
#pragma unroll
    for (int t = 0; t < NTt; t++) acc[t] = 0.f;
    for (int a = 0; a < 7; a++)
      for (int b = 0; b < 7; b++) {
        float xv[NTt];
        const float* tp = &tile[((tu + a) * 22 + (tf + b)) * NTt];
#pragma unroll
        for (int t = 0; t < NTt; t++) xv[t] = tp[t];
#pragma unroll
        for (int c = 0; c < 7; c++) {
          float wv = wl[fil * 343 + a * 49 + b * 7 + c];
#pragma unroll
          for (int t = 0; t < NTt; t++) {
            constexpr int M = NTt;
            int tt = t + c - 3; tt += (tt < 0) ? M : 0; tt -= (tt >= M) ? M : 0;
            acc[t] += xv[tt] * wv;
          }
        }
      }
    long base = (((long)z * NFILq + fil) * NTt) * NUt * NFt + (long)u * NFt + f;
#pragma unroll
    for (int t = 0; t < NTt; t++) sp[base + (long)t * NUt * NFt] = acc[t];
  }
}

// ---------------- 3D circular conv: D (flipped filter, sum over filters) --------
template <int NTt, int NUt, int NFt>
__global__ __launch_bounds__(256) void k_convD(
    const float* __restrict__ sRe, const float* __restrict__ sIm,
    const float* __restrict__ w,
    float* __restrict__ xRe, float* __restrict__ xIm) {
  constexpr int TF = NFt / 16;
  __shared__ float tile[22 * 22 * NTt];
  __shared__ float wl[NFILq * 343];
  int z = blockIdx.z;
  const float* sp = (blockIdx.y ? sIm : sRe) + (long)z * NFILq * NTt * NUt * NFt;
  float* xp = (blockIdx.y ? xIm : xRe) + (long)z * NTt * NUt * NFt;
  int u0 = (blockIdx.x / TF) * 16, f0 = (blockIdx.x % TF) * 16;
  for (int e = threadIdx.x; e < NFILq * 343; e += 256) wl[e] = w[e];
  int tu = threadIdx.x >> 4, tf = threadIdx.x & 15;
  int u = u0 + tu, f = f0 + tf;
  float acc[NTt];
#pragma unroll
  for (int t = 0; t < NTt; t++) acc[t] = 0.f;
  constexpr int TOT = 22 * 22 * NTt;
  for (int fil = 0; fil < NFILq; ++fil) {
    __syncthreads();
    const float* chan = sp + (long)fil * NTt * NUt * NFt;
    for (int e = threadIdx.x; e < TOT; e += 256) {
      int hu = e / (22 * NTt);
      int r = e - hu * 22 * NTt;
      int hf = r / NTt, ht = r - (r / NTt) * NTt;
      int gu = u0 + hu - 3; gu += (gu < 0) ? NUt : 0; gu -= (gu >= NUt) ? NUt : 0;
      int gf = f0 + hf - 3; gf += (gf < 0) ? NFt : 0; gf -= (gf >= NFt) ? NFt : 0;
      tile[e] = chan[((long)ht * NUt + gu) * NFt + gf];
    }
    __syncthreads();
    for (int a = 0; a < 7; a++)
      for (int b = 0; b < 7; b++) {
        float xv[NTt];
        const float* tp = &tile[((tu + a) * 22 + (tf + b)) * NTt];
#pragma unroll
        for (int t = 0; t < NTt; t++) xv[t] = tp[t];
#pragma unroll
        for (int c = 0; c < 7; c++) {
          float wv = wl[fil * 343 + (6 - a) * 49 + (6 - b) * 7 + (6 - c)];
#pragma unroll
          for (int t = 0; t < NTt; t++) {
            constexpr int M = NTt;
            int tt = t + c - 3; tt += (tt < 0) ? M : 0; tt -= (tt >= M) ? M : 0;
            acc[t] += xv[tt] * wv;
          }
        }
      }
  }
  long base = (long)u * NFt + f;
#pragma unroll
  for (int t = 0; t < NTt; t++) xp[base + (long)t * NUt * NFt] = acc[t];
}

// ---------------- elementwise kernels ----------------
__global__ void k_pack(const float* __restrict__ v, float* __restrict__ re,
                       float* __restrict__ im, long n) {
  long i = (long)blockIdx.x * 256 + threadIdx.x;
  if (i < n) { re[i] = v[2 * i]; im[i] = v[2 * i + 1]; }
}
__global__ void k_pack_mask(const float* __restrict__ v, const float* __restrict__ mask,
                            float* __restrict__ re, float* __restrict__ im) {
  long i = (long)blockIdx.x * 256 + threadIdx.x;
  if (i < COILN) {
    float m = mask[i % (IMGN)];
    re[i] = v[2 * i] * m; im[i] = v[2 * i + 1] * m;
  }
}
__global__ void k_apply_mask(float* __restrict__ re, float* __restrict__ im,
                             const float* __restrict__ mask) {
  long i = (long)blockIdx.x * 256 + threadIdx.x;
  if (i < COILN) { float m = mask[i % (IMGN)]; re[i] *= m; im[i] *= m; }
}
__global__ void k_conj_csm_sum(const float* __restrict__ re, const float* __restrict__ im,
                               const float* __restrict__ csm,
                               float* __restrict__ orr, float* __restrict__ oi) {
  long i = (long)blockIdx.x * 256 + threadIdx.x;
  if (i >= IMGN) return;
  int z = (int)(i / ((long)NTq * PLANE));
  long uf = i % PLANE;
  float ar = 0.f, ai = 0.f;
  for (int c = 0; c < NCq; c++) {
    long cix = (((long)c * NZq + z) * PLANE + uf) * 2;
    float sr = csm[cix], si = csm[cix + 1];
    long vi = (long)c * IMGN + i;
    float vr = re[vi], vm = im[vi];
    ar += sr * vr + si * vm;       // conj(csm)*v
    ai += sr * vm - si * vr;
  }
  orr[i] = ar; oi[i] = ai;
}
__global__ void k_mul_csm(const float* __restrict__ xr, const float* __restrict__ xi,
                          const float* __restrict__ csm,
                          float* __restrict__ orr, float* __restrict__ oi) {
  long i = (long)blockIdx.x * 256 + threadIdx.x;
  if (i >= COILN) return;
  long rest = i % IMGN;
  int c = (int)(i / IMGN);
  int z = (int)(rest / ((long)NTq * PLANE));
  long uf = rest % PLANE;
  long cix = (((long)c * NZq + z) * PLANE + uf) * 2;
  float sr = csm[cix], si = csm[cix + 1];
  float vr = xr[rest], vm = xi[rest];
  orr[i] = sr * vr - si * vm;
  oi[i] = sr * vm + si * vr;
}
__global__ void k_rss(const float* __restrict__ re, const float* __restrict__ im,
                      float* __restrict__ out) {
  long i = (long)blockIdx.x * 256 + threadIdx.x;
  if (i >= IMGN) return;
  float acc = 0.f;
  for (int c = 0; c < NCq; c++) {
    long vi = (long)c * IMGN + i;
    float a = re[vi], b = im[vi];
    acc += a * a + b * b;
  }
  out[i] = sqrtf(acc);
}
__global__ void k_interleave(const float* __restrict__ re, const float* __restrict__ im,
                             float* __restrict__ out, long n) {
  long i = (long)blockIdx.x * 256 + threadIdx.x;
  if (i < n) { out[2 * i] = re[i]; out[2 * i + 1] = im[i]; }
}
__global__ void k_shrink_update(float* __restrict__ zr, float* __restrict__ zi,
                                const float* __restrict__ tr, const float* __restrict__ ti,
                                const float* __restrict__ br, const float* __restrict__ bi,
                                float* __restrict__ sr, float* __restrict__ si,
                                const float* __restrict__ scal, float mom, long n) {
  long i = (long)blockIdx.x * 256 + threadIdx.x;
  if (i >= n) return;
  float inv = scal[3], thr = scal[4];
  float gr = zr[i] - (tr[i] - br[i]) * inv;
  float gi = zi[i] - (ti[i] - bi[i]) * inv;
  float mag = sqrtf(gr * gr + gi * gi);
  float fac = fmaxf(mag - thr, 0.f) / fmaxf(mag, 1e-12f);
  float snr = gr * fac, sni = gi * fac;
  float sor = sr[i], soi = si[i];
  zr[i] = snr + mom * (snr - sor);
  zi[i] = sni + mom * (sni - soi);
  sr[i] = snr; si[i] = sni;
}

// ---------------- power-iteration reductions ----------------
__global__ __launch_bounds__(256) void k_pi_rnorm(const float* __restrict__ yr,
                                                  const float* __restrict__ yi,
                                                  int n, float* __restrict__ scal) {
  __shared__ float red[256];
  float s = 0.f;
  for (int i = threadIdx.x; i < n; i += 256) { float a = yr[i], b = yi[i]; s += a * a + b * b; }
  red[threadIdx.x] = s; __syncthreads();
  for (int st = 128; st > 0; st >>= 1) {
    if (threadIdx.x < st) red[threadIdx.x] += red[threadIdx.x + st];
    __syncthreads();
  }
  if (threadIdx.x == 0) scal[0] = rsqrtf(fmaxf(red[0], 1e-30f));
}
__global__ void k_pi_scale(const float* __restrict__ yr, const float* __restrict__ yi,
                           float* __restrict__ xr, float* __restrict__ xi,
                           int n, const float* __restrict__ scal) {
  int i = blockIdx.x * 256 + threadIdx.x;
  if (i < n) { float s = scal[0]; xr[i] = yr[i] * s; xi[i] = yi[i] * s; }
}
__global__ __launch_bounds__(256) void k_pi_finalize(
    const float* __restrict__ xr, const float* __restrict__ xi,
    const float* __restrict__ yr, const float* __restrict__ yi,
    int n, const float* __restrict__ lam, float* __restrict__ scal) {
  __shared__ float r0[256], r1[256];
  float dxy = 0.f, dxx = 0.f;
  for (int i = threadIdx.x; i < n; i += 256) {
    float a = xr[i], b = xi[i], c = yr[i], d = yi[i];
    dxy += a * c + b * d; dxx += a * a + b * b;
  }
  r0[threadIdx.x] = dxy; r1[threadIdx.x] = dxx; __syncthreads();
  for (int st = 128; st > 0; st >>= 1) {
    if (threadIdx.x < st) { r0[threadIdx.x] += r0[threadIdx.x + st]; r1[threadIdx.x] += r1[threadIdx.x + st]; }
    __syncthreads();
  }
  if (threadIdx.x == 0) {
    float opn = fabsf(r0[0] / fmaxf(r1[0], 1e-30f));
    opn = fmaxf(opn, 1e-30f);
    scal[3] = 1.0f / opn;
    scal[4] = expf(lam[0]) / opn;
  }
}

// ---------------- orchestration ----------------
extern "C" void kernel_launch(void* const* d_in, const int* in_sizes, int n_in,
                              void* d_out, int out_size, void* d_ws, size_t ws_size,
                              hipStream_t stream) {
  (void)in_sizes; (void)n_in; (void)out_size; (void)ws_size;
  const float* y    = (const float*)d_in[0];
  const float* csm  = (const float*)d_in[1];
  const float* mask = (const float*)d_in[2];
  const float* filt = (const float*)d_in[3];
  const float* lam  = (const float*)d_in[4];
  const float* s0   = (const float*)d_in[5];
  float* out = (float*)d_out;
  float* ws = (float*)d_ws;

  size_t o = 0;
  float* F_RE = ws + o; o += PLANE;   float* F_IM = ws + o; o += PLANE;
  float* C_RE = ws + o; o += COILN;   float* C_IM = ws + o; o += COILN;
  float* D_RE = ws + o; o += COILN;   float* D_IM = ws + o; o += COILN;
  float* I_RE = ws + o; o += IMGN;    float* I_IM = ws + o; o += IMGN;
  float* AH_RE = ws + o; o += IMGN;   float* AH_IM = ws + o; o += IMGN;
  float* S_RE = ws + o; o += SSPN;    float* S_IM = ws + o; o += SSPN;
  float* Z_RE = ws + o; o += SSPN;    float* Z_IM = ws + o; o += SSPN;
  float* BH_RE = ws + o; o += SSPN;   float* BH_IM = ws + o; o += SSPN;
  float* T_RE = ws + o; o += SSPN;    float* T_IM = ws + o; o += SSPN;
  float* PX_RE = ws + o; o += PINN;   float* PX_IM = ws + o; o += PINN;
  float* PY_RE = ws + o; o += PINN;   float* PY_IM = ws + o; o += PINN;
  float* PD_RE = ws + o; o += PIIMG;  float* PD_IM = ws + o; o += PIIMG;
  float* SCAL = ws + o; o += 8;

  const long PX_OFF = 0, PK_OFF = IMGN, CSM_OFF = PK_OFF + 2 * COILN,
             XRSS_OFF = CSM_OFF + 4718592L;

  // zero FISTA state (S,Z contiguous)
  hipMemsetAsync(S_RE, 0, (size_t)4 * SSPN * sizeof(float), stream);

  dim3 b256(256);
  dim3 b128(128);
  dim3 gDft(144);
  dim3 gCoil((unsigned)((COILN + 255) / 256));
  dim3 gImg((unsigned)((IMGN + 255) / 256));
  dim3 gSsp((unsigned)((SSPN + 255) / 256));
  dim3 gGemm(384 * 12);
  dim3 gConvBig(144, 2, NZq);
  dim3 gConvPi(1, 2, 1);
  dim3 gPin(PINN / 256);

  auto gemmL = [&](const float* br_, const float* bi_, float* cr, float* ci, float sgn) {
    k_cgemm<<<gGemm, b128, 0, stream>>>(F_RE, F_IM, br_, bi_, cr, ci,
                                        0L, (long)PLANE, (long)PLANE, sgn, 1.0f);
  };
  auto gemmR = [&](const float* ar_, const float* ai_, float* cr, float* ci, float sgn) {
    k_cgemm<<<gGemm, b128, 0, stream>>>(ar_, ai_, F_RE, F_IM, cr, ci,
                                        (long)PLANE, 0L, (long)PLANE, 1.0f, sgn);
  };

  // 0) DFT matrix
  k_fill_dft<<<gDft, b256, 0, stream>>>(F_RE, F_IM);

  // 1) xrss = RSS(y): ifft2(y) then RSS
  k_pack<<<gCoil, b256, 0, stream>>>(y, C_RE, C_IM, COILN);
  gemmL(C_RE, C_IM, D_RE, D_IM, -1.0f);
  gemmR(D_RE, D_IM, C_RE, C_IM, -1.0f);
  k_rss<<<gImg, b256, 0, stream>>>(C_RE, C_IM, out + XRSS_OFF);

  // 2) AHy = sum_c conj(csm) * ifft2(mask*y)
  k_pack_mask<<<gCoil, b256, 0, stream>>>(y, mask, C_RE, C_IM);
  gemmL(C_RE, C_IM, D_RE, D_IM, -1.0f);
  gemmR(D_RE, D_IM, C_RE, C_IM, -1.0f);
  k_conj_csm_sum<<<gImg, b256, 0, stream>>>(C_RE, C_IM, csm, AH_RE, AH_IM);

  // 3) BHy = DH(AHy)
  k_convDH<NTq, NUq, NFq><<<gConvBig, b256, 0, stream>>>(AH_RE, AH_IM, filt, BH_RE, BH_IM);

  // 4) power iteration on 16^3 volume: op = DH(D(.))
  k_pack<<<gPin, b256, 0, stream>>>(s0, PX_RE, PX_IM, (long)PINN);
  for (int it = 0; it < 36; ++it) {
    k_convD<16, 16, 16><<<gConvPi, b256, 0, stream>>>(PX_RE, PX_IM, filt, PD_RE, PD_IM);
    k_convDH<16, 16, 16><<<gConvPi, b256, 0, stream>>>(PD_RE, PD_IM, filt, PY_RE, PY_IM);
    k_pi_rnorm<<<1, b256, 0, stream>>>(PY_RE, PY_IM, PINN, SCAL);
    k_pi_scale<<<gPin, b256, 0, stream>>>(PY_RE, PY_IM, PX_RE, PX_IM, PINN, SCAL);
  }
  k_convD<16, 16, 16><<<gConvPi, b256, 0, stream>>>(PX_RE, PX_IM, filt, PD_RE, PD_IM);
  k_convDH<16, 16, 16><<<gConvPi, b256, 0, stream>>>(PD_RE, PD_IM, filt, PY_RE, PY_IM);
  k_pi_finalize<<<1, b256, 0, stream>>>(PX_RE, PX_IM, PY_RE, PY_IM, PINN, lam, SCAL);

  // 5) FISTA: 4 steps
  float beta = 1.0f;
  for (int stp = 0; stp < 4; ++stp) {
    // BHBz = DH( AH( A( D(z) ) ) )
    k_convD<NTq, NUq, NFq><<<gConvBig, b256, 0, stream>>>(Z_RE, Z_IM, filt, I_RE, I_IM);
    k_mul_csm<<<gCoil, b256, 0, stream>>>(I_RE, I_IM, csm, C_RE, C_IM);
    gemmL(C_RE, C_IM, D_RE, D_IM, 1.0f);
    gemmR(D_RE, D_IM, C_RE, C_IM, 1.0f);
    k_apply_mask<<<gCoil, b256, 0, stream>>>(C_RE, C_IM, mask);
    gemmL(C_RE, C_IM, D_RE, D_IM, -1.0f);
    gemmR(D_RE, D_IM, C_RE, C_IM, -1.0f);
    k_conj_csm_sum<<<gImg, b256, 0, stream>>>(C_RE, C_IM, csm, I_RE, I_IM);
    k_convDH<NTq, NUq, NFq><<<gConvBig, b256, 0, stream>>>(I_RE, I_IM, filt, T_RE, T_IM);
    float beta_new = (1.0f + sqrtf(1.0f + 4.0f * beta * beta)) * 0.5f;
    float mom = (beta - 1.0f) / beta_new;
    beta = beta_new;
    k_shrink_update<<<gSsp, b256, 0, stream>>>(Z_RE, Z_IM, T_RE, T_IM, BH_RE, BH_IM,
                                               S_RE, S_IM, SCAL, mom, SSPN);
  }

  // 6) x = D(s); p_k = A(x) (no mask); p_x = RSS(p_k)
  k_convD<NTq, NUq, NFq><<<gConvBig, b256, 0, stream>>>(S_RE, S_IM, filt, I_RE, I_IM);
  k_mul_csm<<<gCoil, b256, 0, stream>>>(I_RE, I_IM, csm, C_RE, C_IM);
  gemmL(C_RE, C_IM, D_RE, D_IM, 1.0f);
  gemmR(D_RE, D_IM, C_RE, C_IM, 1.0f);                       // C = p_k (planar)
  k_interleave<<<gCoil, b256, 0, stream>>>(C_RE, C_IM, out + PK_OFF, COILN);
  gemmL(C_RE, C_IM, D_RE, D_IM, -1.0f);
  gemmR(D_RE, D_IM, C_RE, C_IM, -1.0f);                      // ifft2(p_k)
  k_rss<<<gImg, b256, 0, stream>>>(C_RE, C_IM, out + PX_OFF);

  // 7) csm passthrough output
  hipMemcpyAsync(out + CSM_OFF, csm, 4718592L * sizeof(float),
                 hipMemcpyDeviceToDevice, stream);
}